// MLPAggregator_32959579030026
// MI455X (gfx1250) — compile-verified
//
#include <hip/hip_runtime.h>

typedef __attribute__((ext_vector_type(2))) float v2f;
typedef __attribute__((ext_vector_type(4))) float v4f;
typedef __attribute__((ext_vector_type(8))) float v8f;

// Problem constants (reference: N=100000, K=16, D=128)
constexpr int KNEI       = 16;
constexpr int D          = 128;
constexpr int TWO_D      = 256;
constexpr int LDS_STRIDE = 260;                    // 256 + 4 dword pad -> conflict-free frags
constexpr int W_LDS_DW   = D * LDS_STRIDE;         // 33280 dwords (133,120 B)
constexpr int INP_DW     = 16 * LDS_STRIDE;        // per-wave inp tile, 4160 dwords
constexpr int WAVES      = 8;                      // 256 threads
constexpr int ROWS_PER_BLOCK = WAVES * 16;         // 128
// total dynamic LDS = 33280 + 8*4160 = 66560 dwords = 266,240 B  (<= 320 KB/WGP)

__global__ __launch_bounds__(256)
void mlp_agg_wmma_f32(const float* __restrict__ curr_emb,
                      const float* __restrict__ alpha,
                      const float* __restrict__ msg,
                      const float* __restrict__ W,
                      const float* __restrict__ bias,
                      float* __restrict__ out,
                      int N)
{
    extern __shared__ float smem[];
    float* Wl = smem;                               // W: 128 rows x 256, stride 260
    const int tid  = threadIdx.x;
    const int lane = tid & 31;
    const int wave = tid >> 5;
    const int l16  = lane & 15;
    const int h    = lane >> 4;                     // 0: K even pair, 1: K odd pair
    float* inpL = smem + W_LDS_DW + wave * INP_DW;  // this wave's 16x256 inp tile

    // ---- Phase 0: stage W (128x256 f32) into LDS, coalesced float4 ----
    #pragma unroll 4
    for (int it = 0; it < 32; ++it) {
        int i   = tid + it * 256;                   // float4 index 0..8191
        int e   = i << 2;                           // element index
        int row = e >> 8;                           // / 256
        int col = e & 255;
        v4f w4 = *(const v4f*)(W + e);
        *(v4f*)(Wl + row * LDS_STRIDE + col) = w4;
    }

    // ---- Phase 1: nei_msg = sum_k alpha[n,k]*msg[n,k,:]; inp = [nei | curr] ----
    const int rowBase = blockIdx.x * ROWS_PER_BLOCK + wave * 16;
    for (int r = 0; r < 16; ++r) {
        int n  = rowBase + r;
        int nc = (n < N) ? n : (N - 1);             // clamp so tail block still streams safely
        float av = alpha[(size_t)nc * KNEI + l16];  // lane l16 holds alpha[n][l16]
        const float* mrow = msg + (size_t)nc * (KNEI * D);
        v4f acc = {0.f, 0.f, 0.f, 0.f};
        #pragma unroll
        for (int k = 0; k < KNEI; ++k) {
            float ak = __shfl(av, k, 32);
            v4f m = *(const v4f*)(mrow + k * D + 4 * lane);
            acc.x += ak * m.x; acc.y += ak * m.y;
            acc.z += ak * m.z; acc.w += ak * m.w;
        }
        v4f c4 = *(const v4f*)(curr_emb + (size_t)nc * (KNEI * D) + 4 * lane);
        *(v4f*)(inpL + r * LDS_STRIDE + 4 * lane)       = acc;  // cols 0..127  = nei_msg
        *(v4f*)(inpL + r * LDS_STRIDE + D + 4 * lane)   = c4;   // cols 128..255 = curr
    }
    __syncthreads();   // covers W staging for all waves + own-wave inp ds ordering

    // ---- Phase 2: out[16x128] = inp(16x256) @ W^T via V_WMMA_F32_16X16X4_F32 ----
    // A frag (16x4): lane l<16 -> {inp[l][k0],inp[l][k0+1]}, lanes 16-31 -> k0+2,k0+3
    // B frag (4x16): B[k][nn] = W[16j+nn][k]  -> same float2-from-row pattern
    v8f C[8];
    #pragma unroll
    for (int j = 0; j < 8; ++j) C[j] = (v8f){0,0,0,0,0,0,0,0};

    const float* Arow = inpL + l16 * LDS_STRIDE + 2 * h;
    #pragma unroll
    for (int kcb = 0; kcb < 8; ++kcb) {             // blocks of 8 K-chunks (K-chunk = 4)
        v2f A[8];
        #pragma unroll
        for (int t = 0; t < 8; ++t)
            A[t] = *(const v2f*)(Arow + (kcb * 8 + t) * 4);
        #pragma unroll
        for (int j = 0; j < 8; ++j) {
            const float* Brow = Wl + (16 * j + l16) * LDS_STRIDE + 2 * h;
            #pragma unroll
            for (int t = 0; t < 8; ++t) {
                v2f B = *(const v2f*)(Brow + (kcb * 8 + t) * 4);
                C[j] = __builtin_amdgcn_wmma_f32_16x16x4_f32(
                    /*neg_a=*/false, A[t], /*neg_b=*/false, B,
                    /*c_mod=*/(short)0, C[j], /*reuse_a=*/false, /*reuse_b=*/false);
            }
        }
    }

    // ---- Epilogue: bias + ReLU; C VGPR i: lanes0-15 -> M=i, lanes16-31 -> M=i+8 ----
    #pragma unroll
    for (int j = 0; j < 8; ++j) {
        float bb = bias[16 * j + l16];
        #pragma unroll
        for (int i = 0; i < 8; ++i) {
            int row = rowBase + i + 8 * h;
            if (row < N) {
                float v = C[j][i] + bb;
                out[(size_t)row * D + 16 * j + l16] = (v > 0.f) ? v : 0.f;
            }
        }
    }
}

extern "C" void kernel_launch(void* const* d_in, const int* in_sizes, int n_in,
                              void* d_out, int out_size, void* d_ws, size_t ws_size,
                              hipStream_t stream) {
    const float* curr_emb = (const float*)d_in[0];   // (N,16,128)
    const float* alpha    = (const float*)d_in[1];   // (N,16,1)
    const float* msg      = (const float*)d_in[2];   // (N,16,128)
    const float* W        = (const float*)d_in[3];   // (128,256)
    const float* bias     = (const float*)d_in[4];   // (128,)
    float* out = (float*)d_out;                      // (N,128)

    const int N = in_sizes[0] / (KNEI * D);
    const int blocks = (N + ROWS_PER_BLOCK - 1) / ROWS_PER_BLOCK;
    const size_t lds_bytes = (size_t)(W_LDS_DW + WAVES * INP_DW) * sizeof(float);

    hipLaunchKernelGGL(mlp_agg_wmma_f32, dim3(blocks), dim3(256), lds_bytes, stream,
                       curr_emb, alpha, msg, W, bias, out, N);
}